// WaveRNN_45406394253491
// MI455X (gfx1250) — compile-verified
//
#include <hip/hip_runtime.h>
#include <cstdint>

constexpr int Bn    = 4;
constexpr int Hn    = 256;
constexpr int Wn    = 256;
constexpr int NTn   = 512;
constexpr int NRECn = 128;
constexpr float DTc = 0.001f;
constexpr float DHc = 10.0f;

constexpr int NWG  = 4;                  // workgroups per batch (row split)
constexpr int ROWS = Hn / NWG;           // 64 rows per WG
constexpr int TPB  = 1024;               // threads per WG (32 waves)
constexpr int KC   = (ROWS * Wn) / TPB;  // 16 cells per thread
constexpr int NWG_TOT = Bn * NWG;        // 16 persistent workgroups

// ---------------------------------------------------------------- init ----
__global__ __launch_bounds__(256) void wave_init(unsigned* counter, float* haloG, int nhalo) {
  int i = blockIdx.x * blockDim.x + threadIdx.x;
  if (i == 0) *counter = 0u;
  if (i < nhalo) haloG[i] = 0.0f;
}

// 5-point stencil update for one cell; hn[idx] currently holds h2.
__device__ __forceinline__ float cell_update(const float* __restrict__ h1,
                                             const float* __restrict__ hn,
                                             int idx, int col, float c2) {
  const float c  = h1[idx];
  const float up = h1[idx - Wn];
  const float dn = h1[idx + Wn];
  const float lf = (col > 0)      ? h1[idx - 1] : 0.0f;
  const float rt = (col < Wn - 1) ? h1[idx + 1] : 0.0f;
  const float lap = up + dn + lf + rt - 4.0f * c;
  return 2.0f * c - hn[idx] + c2 * lap;
}

// ---------------------------------------------------- persistent stepper --
__global__ __launch_bounds__(TPB) void wave_step_persistent(
    const float* __restrict__ x, const float* __restrict__ vp,
    const int* __restrict__ src_y, const int* __restrict__ src_x,
    const int* __restrict__ rec_y, const int* __restrict__ rec_x,
    float* __restrict__ out, unsigned* __restrict__ counter,
    float* __restrict__ haloG, float* __restrict__ prePart)
{
  // two ping-pong field buffers, each with 1 halo row above/below: 132 KB LDS
  __shared__ float buf[2][(ROWS + 2) * Wn];

  const int tid = threadIdx.x;
  const int wg  = blockIdx.x;          // 0..15
  const int b   = wg / NWG;            // batch
  const int g   = wg % NWG;            // row-slab within batch
  const int r0  = g * ROWS;            // first global row owned
  const int col = tid & (Wn - 1);      // fixed column per thread
  const int rb  = tid >> 8;            // 0..3 row sub-block (uniform per wave)

  // zero both buffers (h1 = h2 = 0, halo rows = 0 zero-BC)
  {
    float* bb = &buf[0][0];
    for (int i = tid; i < 2 * (ROWS + 2) * Wn; i += TPB) bb[i] = 0.0f;
  }

  // c2/dh^2 and precondition accumulators live in registers
  float c2r[KC], pre[KC];
#pragma unroll
  for (int k = 0; k < KC; ++k) {
    const int gr = r0 + rb + 4 * k;
    const float s = vp[gr * Wn + col] * DTc;
    c2r[k] = (s * s) / (DHc * DHc);
    pre[k] = 0.0f;
  }

  // receiver coordinates cached in registers (one receiver per thread < 128)
  int my_ry = -1, my_rx = 0;
  if (tid < NRECn) {
    my_ry = rec_y[b * NRECn + tid];
    my_rx = rec_x[b * NRECn + tid];
  }
  const int sy = src_y[b], sx = src_x[b];
  const bool own_src = (sy >= r0) && (sy < r0 + ROWS);
  const int  sl = sy - r0;                                   // local source row
  const bool own_cell = own_src && (col == sx) && ((sl & 3) == rb);
  const int  ksrc = sl >> 2;                                 // only used if own_cell

  __syncthreads();

  int cur = 0;  // buf[cur] = h1, buf[cur^1] = h2 (overwritten with hn)
  for (int t = 0; t < NTn; ++t) {
    // ---- 1. ISSUE async halo fill of h1 = buf[cur] (CDNA5 async path,
    //         ASYNCcnt). parity read = (t-1)&1 == (t+1)&1. No wait yet —
    //         latency hidden behind interior compute below.
    if (tid < Wn) {  // 256 lanes, 8 bytes each -> two 1 KB halo rows
      const bool top    = tid < (Wn / 2);
      const int  lane   = tid & (Wn / 2 - 1);           // 0..127
      const bool active = top ? (g > 0) : (g < NWG - 1);
      if (active) {
        const int p    = (t + 1) & 1;
        const int srcg = top ? (g - 1) : (g + 1);
        const int side = top ? 1 : 0;                   // neighbor's bottom / top row
        const unsigned long long gp = (unsigned long long)
            (haloG + ((((size_t)p * Bn + b) * NWG + srcg) * 2 + side) * Wn + lane * 2);
        const int ldr = top ? 0 : (ROWS + 1);
        const uint32_t lds = (uint32_t)(uint64_t)(&buf[cur][ldr * Wn + lane * 2]);
        asm volatile("global_load_async_to_lds_b64 %0, %1, off"
                     :: "v"(lds), "v"(gp) : "memory");
      }
    }

    const float* __restrict__ h1 = buf[cur];
    float* __restrict__ hn = buf[cur ^ 1];
    const float xt = own_cell ? x[b * NTn + t] : 0.0f;  // source sample (1 lane)

    // ---- 2. interior cells (halo-independent): overlap with async halo fill
    float vals[KC];
#pragma unroll
    for (int k = 0; k < KC; ++k) {
      const bool boundary = (rb == 0 && k == 0) || (rb == 3 && k == KC - 1);
      if (!boundary) {
        const int idx = (1 + rb + 4 * k) * Wn + col;
        float v = cell_update(h1, hn, idx, col, c2r[k]);
        if (own_cell && k == ksrc) v += xt;             // fold source injection
        hn[idx] = v;
        vals[k] = v;
      }
    }

    // ---- 3. wait for async halos (each issuing wave waits its own count),
    //         then barrier so all waves see the halo rows in LDS
    if (tid < Wn) asm volatile("s_wait_asynccnt 0x0" ::: "memory");
    __syncthreads();

    // ---- 4. boundary-adjacent rows (lr==1 uses halo row 0; lr==ROWS uses
    //         halo row ROWS+1)
    if (rb == 0) {
      const int idx = 1 * Wn + col;
      float v = cell_update(h1, hn, idx, col, c2r[0]);
      if (own_cell && 0 == ksrc) v += xt;
      hn[idx] = v;
      vals[0] = v;
    } else if (rb == 3) {
      const int idx = ROWS * Wn + col;
      float v = cell_update(h1, hn, idx, col, c2r[KC - 1]);
      if (own_cell && (KC - 1) == ksrc) v += xt;
      hn[idx] = v;
      vals[KC - 1] = v;
    }
    __syncthreads();

    // ---- 5. receiver sampling (post-injection values, matches reference)
    if (my_ry >= r0 && my_ry < r0 + ROWS)
      out[(size_t)t * (Bn * NRECn) + b * NRECn + tid] =
          hn[(my_ry - r0 + 1) * Wn + my_rx];

    // ---- 6. precondition accumulation straight from registers
#pragma unroll
    for (int k = 0; k < KC; ++k) pre[k] += vals[k] * vals[k];

    // ---- 7. publish boundary rows from registers, parity write = t&1
    {
      const size_t hb = ((((size_t)(t & 1) * Bn + b) * NWG + g) * 2) * Wn;
      if (rb == 0) haloG[hb + col]      = vals[0];       // our top row (lr=1)
      if (rb == 3) haloG[hb + Wn + col] = vals[KC - 1];  // our bottom row (lr=ROWS)
    }

    // ---- 8. device-wide barrier (monotonic counter, release/acquire)
    __threadfence();          // each wave drains its halo/receiver stores
    __syncthreads();
    if (tid == 0) {
      __hip_atomic_fetch_add(counter, 1u, __ATOMIC_RELEASE, __HIP_MEMORY_SCOPE_AGENT);
      const unsigned target = (unsigned)(t + 1) * (unsigned)NWG_TOT;
      while (__hip_atomic_load(counter, __ATOMIC_ACQUIRE, __HIP_MEMORY_SCOPE_AGENT) < target)
        __builtin_amdgcn_s_sleep(1);
    }
    __syncthreads();

    cur ^= 1;
  }

  // per-batch precondition partials (disjoint rows -> deterministic)
#pragma unroll
  for (int k = 0; k < KC; ++k)
    prePart[((size_t)b * Hn + (r0 + rb + 4 * k)) * Wn + col] = pre[k];
}

// ------------------------------------------------------------- reduce -----
__global__ __launch_bounds__(256) void pre_reduce(const float* __restrict__ prePart,
                                                  float* __restrict__ out) {
  const int i = blockIdx.x * blockDim.x + threadIdx.x;  // 0..H*W-1
  out[(size_t)NTn * Bn * NRECn + i] =
      prePart[i] + prePart[Hn * Wn + i] + prePart[2 * Hn * Wn + i] + prePart[3 * Hn * Wn + i];
}

// ------------------------------------------------------------- launch -----
extern "C" void kernel_launch(void* const* d_in, const int* in_sizes, int n_in,
                              void* d_out, int out_size, void* d_ws, size_t ws_size,
                              hipStream_t stream) {
  (void)in_sizes; (void)n_in; (void)out_size; (void)ws_size;
  const float* x     = (const float*)d_in[0];
  const float* vp    = (const float*)d_in[1];
  const int*   src_y = (const int*)d_in[2];
  const int*   src_x = (const int*)d_in[3];
  const int*   rec_y = (const int*)d_in[4];
  const int*   rec_x = (const int*)d_in[5];
  float* out = (float*)d_out;

  // workspace layout: [counter pad 256B][halo 2*B*NWG*2*W floats][prePart B*H*W floats]
  unsigned* counter = (unsigned*)d_ws;
  float* haloG = (float*)((char*)d_ws + 256);
  const int nhalo = 2 * Bn * NWG * 2 * Wn;  // 16384 floats
  float* prePart = (float*)((char*)d_ws + 256 + (size_t)nhalo * sizeof(float));

  wave_init<<<(nhalo + 255) / 256, 256, 0, stream>>>(counter, haloG, nhalo);
  wave_step_persistent<<<NWG_TOT, TPB, 0, stream>>>(x, vp, src_y, src_x, rec_y, rec_x,
                                                    out, counter, haloG, prePart);
  pre_reduce<<<(Hn * Wn) / 256, 256, 0, stream>>>(prePart, out);
}